// GCNBranch_neg_Normal_A_34437047780015
// MI455X (gfx1250) — compile-verified
//
#include <hip/hip_runtime.h>

typedef float v2f __attribute__((ext_vector_type(2)));
typedef float v8f __attribute__((ext_vector_type(8)));

#define IN_DIM 512
#define H1     256
#define H2T    62     // true hidden-2 width
#define H2P    64     // padded to multiple of 16/64
#define OD     128

// ---------------------------------------------------------------------------
// WMMA f32 GEMM:  C[N,Fout] = A[N,K] @ W[K,Fout] (+ bias), cols >= trueCols
// forced to zero (used to keep the 62->64 padding exactly zero).
// One wave computes a 16x64 strip: A-frag reused over 4 column tiles.
// Uses V_WMMA_F32_16X16X4_F32 (native fp32 matrix op, wave32).
// ---------------------------------------------------------------------------
__global__ __launch_bounds__(128) void gemm_wmma_k(
    const float* __restrict__ A, const float* __restrict__ W,
    const float* __restrict__ bias, float* __restrict__ C,
    int Nrows, int K, int Fout, int trueCols)
{
    const int lane = threadIdx.x & 31;
    const int wid  = threadIdx.x >> 5;
    const int colGroups = Fout >> 6;              // 64-wide column groups
    const int rowTiles  = (Nrows + 15) >> 4;
    const int total     = rowTiles * colGroups;
    const int t = blockIdx.x * 4 + wid;           // wave-uniform -> EXEC stays full
    if (t >= total) return;

    const int rowTile = t / colGroups;
    const int cg      = t - rowTile * colGroups;
    const int rowBase = rowTile << 4;
    const int colBase = cg << 6;

    // A fragment addressing (16x4 f32 layout):
    //   lanes 0-15: M=lane, holds K = k+0 (v0), k+1 (v1)
    //   lanes16-31: M=lane-16, holds K = k+2 (v0), k+3 (v1)
    const int mrow = lane & 15;
    const int koff = (lane >> 4) << 1;            // 0 or 2
    int ra = rowBase + mrow;
    if (ra > Nrows - 1) ra = Nrows - 1;           // safe clamp (N%16==0 normally)
    const float* Ap = A + (size_t)ra * K + koff;
    const int n = lane & 15;                      // B/C column within 16-tile

    v8f c0 = {0.f,0.f,0.f,0.f,0.f,0.f,0.f,0.f};
    v8f c1 = c0, c2 = c0, c3 = c0;

    for (int k = 0; k < K; k += 4) {
        v2f a;
        a.x = Ap[k];
        a.y = Ap[k + 1];
        // B fragment (4x16 f32): v0 = row (k+koff), v1 = row (k+koff+1)
        const float* W0 = W + (size_t)(k + koff) * Fout + colBase + n;
        const float* W1 = W0 + Fout;
        v2f b0, b1, b2, b3;
        b0.x = W0[0];  b0.y = W1[0];
        b1.x = W0[16]; b1.y = W1[16];
        b2.x = W0[32]; b2.y = W1[32];
        b3.x = W0[48]; b3.y = W1[48];
        c0 = __builtin_amdgcn_wmma_f32_16x16x4_f32(false, a, false, b0, (short)0, c0, false, false);
        c1 = __builtin_amdgcn_wmma_f32_16x16x4_f32(false, a, false, b1, (short)0, c1, false, false);
        c2 = __builtin_amdgcn_wmma_f32_16x16x4_f32(false, a, false, b2, (short)0, c2, false, false);
        c3 = __builtin_amdgcn_wmma_f32_16x16x4_f32(false, a, false, b3, (short)0, c3, false, false);
    }

    // C/D layout: VGPR r -> M = r (lanes 0-15) / M = 8+r (lanes 16-31), N = lane&15
    const int mBase = (lane >> 4) << 3;           // 0 or 8
#pragma unroll
    for (int r = 0; r < 8; ++r) {
        const int row = rowBase + mBase + r;
        if (row >= Nrows) continue;
        float vals[4] = { c0[r], c1[r], c2[r], c3[r] };
#pragma unroll
        for (int s = 0; s < 4; ++s) {
            const int col = colBase + s * 16 + n;
            float v = vals[s];
            float bv = 0.f;
            if (bias != nullptr && col < trueCols) bv = bias[col];
            C[(size_t)row * Fout + col] = (col < trueCols) ? (v + bv) : 0.0f;
        }
    }
}

// ---------------------------------------------------------------------------
// Degree / normalization (self-loops folded in: deg starts at 1.0)
// ---------------------------------------------------------------------------
__global__ void deg_init_k(float* deg, int n) {
    int i = blockIdx.x * blockDim.x + threadIdx.x;
    if (i < n) deg[i] = 1.0f;
}
__global__ void deg_accum_k(const int* __restrict__ dst, float* deg, int E) {
    int i = blockIdx.x * blockDim.x + threadIdx.x;
    if (i < E) atomicAdd(&deg[dst[i]], 1.0f);
}
__global__ void dinv_k(const float* __restrict__ deg, float* dinv, int n) {
    int i = blockIdx.x * blockDim.x + threadIdx.x;
    if (i < n) dinv[i] = rsqrtf(deg[i]);          // deg >= 1 always (self loop)
}

// ---------------------------------------------------------------------------
// Zero-padding copies for the 62-wide layer
// ---------------------------------------------------------------------------
__global__ void pad_mat_k(const float* __restrict__ s, float* d,
                          int sr, int sc, int dr, int dc) {
    int i = blockIdx.x * blockDim.x + threadIdx.x;
    if (i >= dr * dc) return;
    int r = i / dc, c = i - r * dc;
    d[i] = (r < sr && c < sc) ? s[r * sc + c] : 0.0f;
}
__global__ void pad_vec_k(const float* __restrict__ s, float* d, int sn, int dn) {
    int i = blockIdx.x * blockDim.x + threadIdx.x;
    if (i < dn) d[i] = (i < sn) ? s[i] : 0.0f;
}

// ---------------------------------------------------------------------------
// Aggregation: self-loop init then edge scatter (L2-resident float atomics)
// ---------------------------------------------------------------------------
__global__ void selfloop_k(const float* __restrict__ h, const float* __restrict__ dinv,
                           float* __restrict__ agg, long long nf, int logF) {
    long long i = (long long)blockIdx.x * blockDim.x + threadIdx.x;
    if (i >= nf) return;
    int node = (int)(i >> logF);
    float d = dinv[node];
    agg[i] = h[i] * d * d;
}
__global__ void edge_msg_k(const int* __restrict__ src, const int* __restrict__ dst,
                           const float* __restrict__ dinv, const float* __restrict__ h,
                           float* __restrict__ agg, long long ef, int logF) {
    long long i = (long long)blockIdx.x * blockDim.x + threadIdx.x;
    if (i >= ef) return;
    int e = (int)(i >> logF);
    int f = (int)(i & ((1 << logF) - 1));
    int s = src[e], d = dst[e];
    float w = dinv[s] * dinv[d];
    atomicAdd(&agg[((long long)d << logF) + f], h[((long long)s << logF) + f] * w);
}

// out = base + scale * (relu? max(agg+bias,0) : agg+bias)
__global__ void combine_k(const float* __restrict__ base, const float* __restrict__ agg,
                          const float* __restrict__ bias, float* __restrict__ out,
                          long long n, int logF, float scale, int do_relu) {
    long long i = (long long)blockIdx.x * blockDim.x + threadIdx.x;
    if (i >= n) return;
    int f = (int)(i & ((1 << logF) - 1));
    float v = agg[i] + bias[f];
    if (do_relu) v = fmaxf(v, 0.0f);
    out[i] = base[i] + scale * v;
}

// ---------------------------------------------------------------------------
extern "C" void kernel_launch(void* const* d_in, const int* in_sizes, int n_in,
                              void* d_out, int out_size, void* d_ws, size_t ws_size,
                              hipStream_t stream)
{
    const float* x   = (const float*)d_in[0];
    const int*   ei  = (const int*)d_in[1];        // [2,E] flat: row0=src, row1=dst
    const float* W1  = (const float*)d_in[2];
    const float* b1  = (const float*)d_in[3];
    const float* W2  = (const float*)d_in[4];
    const float* b2  = (const float*)d_in[5];
    const float* W3  = (const float*)d_in[6];
    const float* b3  = (const float*)d_in[7];
    const float* Wg1 = (const float*)d_in[8];
    const float* bg1 = (const float*)d_in[9];
    const float* Wg2 = (const float*)d_in[10];
    const float* bg2 = (const float*)d_in[11];
    const float* Wg3 = (const float*)d_in[12];
    const float* bg3 = (const float*)d_in[13];
    const float* Wg4 = (const float*)d_in[14];
    const float* bg4 = (const float*)d_in[15];
    const float* Wg5 = (const float*)d_in[16];
    const float* bg5 = (const float*)d_in[17];
    const float* Wg6 = (const float*)d_in[18];
    const float* bg6 = (const float*)d_in[19];

    const int N = in_sizes[0] / IN_DIM;
    const int E = in_sizes[1] / 2;
    const int* srcIdx = ei;
    const int* dstIdx = ei + E;

    // ---- workspace carve-up (floats) ----
    float* ws   = (float*)d_ws;
    float* deg  = ws; ws += N;
    float* dinv = ws; ws += N;
    float* Wg2p = ws; ws += H2P * H2P;
    float* bg2p = ws; ws += H2P;
    float* W3p  = ws; ws += H2P * OD;
    const size_t big = (size_t)N * H1;
    float* bufA = ws; ws += big;                   // x1l / x2l / x3l / x4 ...
    float* bufB = ws; ws += big;                   // h
    float* bufC = ws; ws += big;                   // agg
    float* bufD = ws; ws += big;                   // x1 / x2 / x3 / x5

    auto gemm = [&](const float* A, const float* W, const float* bias, float* C,
                    int K, int Fout, int trueCols) {
        int tiles  = ((N + 15) / 16) * (Fout / 64);
        int blocks = (tiles + 3) / 4;
        gemm_wmma_k<<<blocks, 128, 0, stream>>>(A, W, bias, C, N, K, Fout, trueCols);
    };
    auto aggregate = [&](const float* h, float* agg, int F, int logF) {
        long long nf = (long long)N * F;
        selfloop_k<<<(unsigned)((nf + 255) / 256), 256, 0, stream>>>(h, dinv, agg, nf, logF);
        long long ef = (long long)E * F;
        edge_msg_k<<<(unsigned)((ef + 255) / 256), 256, 0, stream>>>(srcIdx, dstIdx, dinv, h, agg, ef, logF);
    };
    auto combine = [&](const float* base, const float* agg, const float* bias, float* out,
                       int F, int logF, float scale, int relu) {
        long long n = (long long)N * F;
        combine_k<<<(unsigned)((n + 255) / 256), 256, 0, stream>>>(base, agg, bias, out, n, logF, scale, relu);
    };

    // ---- normalization (D^-1/2 A_hat D^-1/2) ----
    deg_init_k<<<(N + 255) / 256, 256, 0, stream>>>(deg, N);
    deg_accum_k<<<(E + 255) / 256, 256, 0, stream>>>(dstIdx, deg, E);
    dinv_k<<<(N + 255) / 256, 256, 0, stream>>>(deg, dinv, N);

    // ---- pad 62-wide params to 64 ----
    pad_mat_k<<<(H2P * H2P + 255) / 256, 256, 0, stream>>>(Wg2, Wg2p, H2T, H2T, H2P, H2P);
    pad_vec_k<<<1, 256, 0, stream>>>(bg2, bg2p, H2T, H2P);
    pad_mat_k<<<(H2P * OD + 255) / 256, 256, 0, stream>>>(W3, W3p, H2T, OD, H2P, OD);

    // ---- layer 1: x1l = x@W1+b1 ; x1 = x1l + relu(gcn(x1l,Wg1,bg1)) ----
    gemm(x, W1, b1, bufA, IN_DIM, H1, H1);
    gemm(bufA, Wg1, nullptr, bufB, H1, H1, H1);
    aggregate(bufB, bufC, H1, 8);
    combine(bufA, bufC, bg1, bufD, H1, 8, 1.0f, 1);          // x1 -> bufD

    // ---- layer 2 (padded 62->64) ----
    gemm(bufD, W2, b2, bufA, H1, H2P, H2T);                  // x2l (cols 62,63 = 0)
    gemm(bufA, Wg2p, nullptr, bufB, H2P, H2P, H2P);
    aggregate(bufB, bufC, H2P, 6);
    combine(bufA, bufC, bg2p, bufD, H2P, 6, 1.0f, 1);        // x2 -> bufD

    // ---- layer 3: x3l = x2@W3+b3 ; x3 = x3l + 0.5*relu(gcn) ----
    gemm(bufD, W3p, b3, bufA, H2P, OD, OD);                  // x3l -> bufA
    gemm(bufA, Wg3, nullptr, bufB, OD, OD, OD);
    aggregate(bufB, bufC, OD, 7);
    combine(bufA, bufC, bg3, bufD, OD, 7, 0.5f, 1);          // x3 -> bufD

    // ---- layer 4: x4 = x3 + 0.5*relu(gcn(x3,Wg4)) ----
    gemm(bufD, Wg4, nullptr, bufB, OD, OD, OD);
    aggregate(bufB, bufC, OD, 7);
    combine(bufD, bufC, bg4, bufA, OD, 7, 0.5f, 1);          // x4 -> bufA

    // ---- layer 5: x5 = x4 + 0.25*relu(gcn(x4,Wg5)) ----
    gemm(bufA, Wg5, nullptr, bufB, OD, OD, OD);
    aggregate(bufB, bufC, OD, 7);
    combine(bufA, bufC, bg5, bufD, OD, 7, 0.25f, 1);         // x5 -> bufD

    // ---- layer 6: x6 = x5 + 0.25*gcn(x5,Wg6)  (no relu) ----
    gemm(bufD, Wg6, nullptr, bufB, OD, OD, OD);
    aggregate(bufB, bufC, OD, 7);
    combine(bufD, bufC, bg6, (float*)d_out, OD, 7, 0.25f, 0); // x6 -> d_out
}